// AggregationHyperbolic_70480413327938
// MI455X (gfx1250) — compile-verified
//
#include <hip/hip_runtime.h>
#include <math.h>

// CDNA5 / gfx1250, wave32. Exact-fp32 matrix path via V_WMMA_F32_16X16X4_F32,
// async global->LDS staging (ASYNCcnt), deterministic split-K.

typedef float v2f __attribute__((ext_vector_type(2)));
typedef float v8f __attribute__((ext_vector_type(8)));
typedef int   v4i __attribute__((ext_vector_type(4)));

#define N_ROWS 8192
#define DIM 64
#define KTILE 32
#define ROWTILE 64
#define SPLITK 2
#define KSPAN (N_ROWS / SPLITK)
#define A_STRIDE 36   // 32 + pad; 144B row pitch (16B aligned), spreads LDS banks
#define B_STRIDE 68   // 64 + pad; 272B row pitch (16B aligned), spreads LDS banks

#if __has_builtin(__builtin_amdgcn_global_load_async_to_lds_b128)
#define HAVE_ASYNC_LDS 1
#endif

__device__ __forceinline__ void async_copy_b128(const float* gsrc, float* ldst) {
#ifdef HAVE_ASYNC_LDS
  __builtin_amdgcn_global_load_async_to_lds_b128(
      (__attribute__((address_space(1))) v4i*)gsrc,
      (__attribute__((address_space(3))) v4i*)ldst, 0, 0);
#else
  *(float4*)ldst = *(const float4*)gsrc;  // fallback: VGPR staging
#endif
}

__device__ __forceinline__ void wait_async_copies() {
#ifdef HAVE_ASYNC_LDS
#if __has_builtin(__builtin_amdgcn_s_wait_asynccnt)
  __builtin_amdgcn_s_wait_asynccnt(0);
#else
  asm volatile("s_wait_asynccnt 0x0" ::: "memory");
#endif
#endif
}

__device__ __forceinline__ float wave_sum(float v) {
#pragma unroll
  for (int m = 16; m >= 1; m >>= 1) v += __shfl_xor(v, m, 32);
  return v;
}

__device__ __forceinline__ float artanh_c(float z) {
  z = fminf(fmaxf(z, -1.0f + 1e-7f), 1.0f - 1e-7f);
  return 0.5f * logf((1.0f + z) / (1.0f - z));
}

// ---------------- Kernel 1: xt = logmap0(x, c); xn = ||xt|| ----------------
__global__ __launch_bounds__(256) void logmap0_kernel(
    const float* __restrict__ x, const float* __restrict__ cptr,
    float* __restrict__ xt, float* __restrict__ xn_out) {
  const int row  = (blockIdx.x * blockDim.x + threadIdx.x) >> 5;
  const int lane = threadIdx.x & 31;
  const float sc = sqrtf(cptr[0]);
  float2 v = ((const float2*)(x + (size_t)row * DIM))[lane];
  float ss = wave_sum(v.x * v.x + v.y * v.y);
  float n  = fmaxf(sqrtf(ss), 1e-15f);
  float f  = artanh_c(sc * n) / (sc * n);
  float2 o; o.x = v.x * f; o.y = v.y * f;
  ((float2*)(xt + (size_t)row * DIM))[lane] = o;
  if (lane == 0) xn_out[row] = fmaxf(f * n, 1e-15f);
}

// ---------------- Kernel 2: partial mx = adj[:, ksplit] @ xt[ksplit] -------
// grid (128, SPLITK); block = 256 threads = 8 waves; block tile 64 rows x 64 cols.
// wave: rowPair = wave>>2 (two 16-row accumulators), colT = (wave&3)*16.
__global__ __launch_bounds__(256) void gemm_kernel(
    const float* __restrict__ adj, const float* __restrict__ xt,
    float* __restrict__ out0, float* __restrict__ out1) {
  __shared__ __align__(16) float As[2][ROWTILE * A_STRIDE];  // 18432 B
  __shared__ __align__(16) float Bs[2][KTILE * B_STRIDE];    // 17408 B

  const int t       = threadIdx.x;
  const int wave    = t >> 5;
  const int lane    = t & 31;
  const int half    = lane >> 4;        // K-half selector per ISA A/B layout
  const int l16     = lane & 15;
  const int rowPair = wave >> 2;        // 0..1 -> rows rowPair*32 + {0..31}
  const int colT    = (wave & 3) * 16;  // 0,16,32,48
  const size_t rowBase = (size_t)blockIdx.x * ROWTILE;
  const int kb0   = blockIdx.y * KSPAN;
  const int kbEnd = kb0 + KSPAN;
  float* __restrict__ dst = blockIdx.y ? out1 : out0;

  v8f acc0 = {};
  v8f acc1 = {};

  // stage one K-tile (A: 64x32, B: 32x64) -> LDS buffer `buf`, async
  auto stage = [&](int kb, int buf) {
#pragma unroll
    for (int i = 0; i < 2; ++i) {
      int lin = t + i * 256;                 // 0..511
      int r = lin >> 3, c4 = (lin & 7) << 2; // A: 8 float4 per row
      async_copy_b128(adj + (rowBase + r) * N_ROWS + kb + c4,
                      &As[buf][r * A_STRIDE + c4]);
    }
#pragma unroll
    for (int i = 0; i < 2; ++i) {
      int lin = t + i * 256;
      int rb = lin >> 4, cb = (lin & 15) << 2;  // B: 16 float4 per row
      async_copy_b128(xt + (size_t)(kb + rb) * DIM + cb,
                      &Bs[buf][rb * B_STRIDE + cb]);
    }
  };

  stage(kb0, 0);
  wait_async_copies();
  __syncthreads();

  int buf = 0;
  for (int kb = kb0; kb < kbEnd; kb += KTILE) {
    const int nkb = kb + KTILE;
    if (nkb < kbEnd) {
      stage(nkb, buf ^ 1);
      if (nkb + KTILE < kbEnd)  // L2 prefetch two K-tiles ahead
        __builtin_prefetch(adj + (rowBase + (t & 63)) * N_ROWS + nkb + KTILE, 0, 3);
    }

    const float* A = As[buf];
    const float* B = Bs[buf];
#pragma unroll
    for (int k = 0; k < KTILE; k += 4) {
      // A 16x4 frags: lanes 0-15 hold (K=k,k+1), lanes 16-31 hold (K=k+2,k+3)
      v2f a0 = *(const v2f*)(A + (rowPair * 32 + l16) * A_STRIDE + k + half * 2);
      v2f a1 = *(const v2f*)(A + (rowPair * 32 + 16 + l16) * A_STRIDE + k + half * 2);
      // B 4x16 frag (shared by both row bands)
      v2f b;
      b.x = B[(k + half * 2 + 0) * B_STRIDE + colT + l16];
      b.y = B[(k + half * 2 + 1) * B_STRIDE + colT + l16];
      acc0 = __builtin_amdgcn_wmma_f32_16x16x4_f32(
          false, a0, false, b, (short)0, acc0, false, false);
      acc1 = __builtin_amdgcn_wmma_f32_16x16x4_f32(
          false, a1, false, b, (short)0, acc1, false, false);
    }
    wait_async_copies();   // our copies into buf^1 have landed in LDS
    __syncthreads();       // everyone's copies landed / everyone done reading
    buf ^= 1;
  }

  // C/D layout: VGPR j -> M = j (lanes 0-15) / j+8 (lanes 16-31), N = lane%16
#pragma unroll
  for (int j = 0; j < 8; ++j) {
    size_t r0 = rowBase + (size_t)rowPair * 32 + half * 8 + j;
    dst[r0 * DIM + colT + l16]        = acc0[j];
    dst[(r0 + 16) * DIM + colT + l16] = acc1[j];
  }
}

// ---------------- Kernel 3: sum split-K + mobius scale + expmap0 + proj ----
__global__ __launch_bounds__(256) void epilogue_kernel(
    float* __restrict__ out, const float* __restrict__ part1,
    const float* __restrict__ xn_arr, const float* __restrict__ cptr) {
  const int row  = (blockIdx.x * blockDim.x + threadIdx.x) >> 5;
  const int lane = threadIdx.x & 31;
  const float sc = sqrtf(cptr[0]);
  float2 m0 = ((const float2*)(out   + (size_t)row * DIM))[lane];
  float2 m1 = ((const float2*)(part1 + (size_t)row * DIM))[lane];
  float2 m; m.x = m0.x + m1.x; m.y = m0.y + m1.y;   // deterministic combine
  float ss = wave_sum(m.x * m.x + m.y * m.y);
  const bool zero_row = (ss == 0.0f);               // all(mx==0) <=> sum sq == 0
  float mxn = fmaxf(sqrtf(ss), 1e-15f);
  float xn  = xn_arr[row];
  float tt  = tanhf(mxn / xn * artanh_c(sc * xn)) / (mxn * sc);
  float rx = m.x * tt, ry = m.y * tt;               // res = support_t
  // expmap0
  float un = fmaxf(sqrtf(wave_sum(rx * rx + ry * ry)), 1e-15f);
  float g  = tanhf(sc * un) / (sc * un);
  float ox = rx * g, oy = ry * g;
  // proj
  float on   = fmaxf(sqrtf(wave_sum(ox * ox + oy * oy)), 1e-15f);
  float maxn = (1.0f - 1e-5f) / sc;
  if (on > maxn) { float s = maxn / on; ox *= s; oy *= s; }
  if (zero_row) { ox = 0.0f; oy = 0.0f; }
  float2 o; o.x = ox; o.y = oy;
  ((float2*)(out + (size_t)row * DIM))[lane] = o;
}

extern "C" void kernel_launch(void* const* d_in, const int* in_sizes, int n_in,
                              void* d_out, int out_size, void* d_ws, size_t ws_size,
                              hipStream_t stream) {
  const float* x   = (const float*)d_in[0];
  const float* adj = (const float*)d_in[1];
  const float* c   = (const float*)d_in[2];
  float* out   = (float*)d_out;
  float* xt    = (float*)d_ws;                        // 8192*64 f32
  float* xn    = xt + (size_t)N_ROWS * DIM;           // + 8192 f32
  float* part1 = xn + N_ROWS;                         // + 8192*64 f32

  logmap0_kernel <<<N_ROWS / 8, 256, 0, stream>>>(x, c, xt, xn);
  gemm_kernel    <<<dim3(N_ROWS / ROWTILE, SPLITK), 256, 0, stream>>>(adj, xt, out, part1);
  epilogue_kernel<<<N_ROWS / 8, 256, 0, stream>>>(out, part1, xn, c);
}